// HierarchicalAttentionPooling_53807350284431
// MI455X (gfx1250) — compile-verified
//
#include <hip/hip_runtime.h>
#include <hip/hip_bf16.h>
#include <stdint.h>

#define G_   256
#define N_   1024
#define C_   128
#define DEG_ 16
#define K_   512
#define TOTAL_ (G_ * N_)
#define E_     (TOTAL_ * DEG_)

typedef __attribute__((ext_vector_type(2))) float v2f;
typedef __attribute__((ext_vector_type(8))) float v8f;

// ---------------------------------------------------------------------------
// Kernel 1: node scores = sigmoid(x . w_score + b); one wave32 per node.
// Also initializes node_id_map to -1 (fresh every call).
// ---------------------------------------------------------------------------
__global__ void hap_score_kernel(const float* __restrict__ x,
                                 const float* __restrict__ w_score,
                                 const float* __restrict__ b_score,
                                 float* __restrict__ scores,
                                 int* __restrict__ node_map) {
    const int wave = threadIdx.x >> 5;
    const int lane = threadIdx.x & 31;
    const int node = blockIdx.x * 8 + wave;

    const float4 xv = ((const float4*)(x + (size_t)node * C_))[lane];
    const float4 wv = ((const float4*)w_score)[lane];
    float s = xv.x * wv.x + xv.y * wv.y + xv.z * wv.z + xv.w * wv.w;
    // wave32 butterfly reduction
    for (int off = 16; off > 0; off >>= 1) s += __shfl_xor(s, off, 32);

    if (lane == 0) {
        const float z = s + b_score[0];
        scores[node]   = 1.0f / (1.0f + __expf(-z));
        node_map[node] = -1;
    }
}

// ---------------------------------------------------------------------------
// Kernel 2: per-graph descending bitonic sort of (score, index) keys in LDS.
// 512 threads sort 1024 keys; rank r < K selects node, builds node_id_map,
// and writes new_batch. Key = (score_bits << 32) | (0xFFFFFFFF - idx):
// sigmoid output is positive so float bits are order-preserving, and the
// secondary key reproduces top_k's lower-index-first tie-breaking.
// ---------------------------------------------------------------------------
__global__ void hap_topk_kernel(const float* __restrict__ scores,
                                int* __restrict__ selected,
                                int* __restrict__ node_map,
                                float* __restrict__ out_batch) {
    __shared__ uint64_t keys[N_];
    const unsigned g = blockIdx.x;
    const unsigned t = threadIdx.x;   // 0..511

    for (unsigned i = t; i < N_; i += 512u) {
        const uint32_t sb = __float_as_uint(scores[g * N_ + i]);
        keys[i] = ((uint64_t)sb << 32) | (uint64_t)(0xFFFFFFFFu - i);
    }
    __syncthreads();

    for (unsigned k = 2; k <= N_; k <<= 1) {
        for (unsigned j = k >> 1; j > 0; j >>= 1) {
            const unsigned i = 2u * t - (t & (j - 1u));
            const unsigned p = i ^ j;
            const uint64_t a = keys[i];
            const uint64_t b = keys[p];
            const bool up = ((i & k) == 0u);
            const bool sw = up ? (a < b) : (a > b);   // final order: descending
            if (sw) { keys[i] = b; keys[p] = a; }
            __syncthreads();
        }
    }

    // keys[0..1023] now descending; rank t < K_ is selected.
    const uint64_t kv  = keys[t];
    const int idx      = (int)(0xFFFFFFFFu - (uint32_t)kv);
    const int new_id   = (int)(g * K_ + t);
    selected[new_id]          = (int)(g * N_) + idx;
    node_map[g * N_ + idx]    = new_id;
    out_batch[new_id]         = (float)g;
}

// ---------------------------------------------------------------------------
// Kernel 3: pooled_x = x[selected] @ w_t + b_t via V_WMMA_F32_16X16X4_F32.
// Block = 256 threads (8 waves). Block owns 16 gathered rows (staged in LDS,
// stride padded to 132 floats -> conflict-free A-fragment reads). Wave w owns
// output columns [16w, 16w+16). K = 128 -> 32 WMMA steps of depth 4.
//
// A fragment (16x4 f32, ISA layout): lane L holds row M = L%16,
//   VGPR0/1 = K = base + 2*(L/16) + {0,1}.
// C/D layout: VGPR r -> M = r + 8*(L/16), N = L%16.
// ---------------------------------------------------------------------------
__global__ void hap_gemm_kernel(const float* __restrict__ x,
                                const float* __restrict__ w_t,
                                const float* __restrict__ b_t,
                                const int* __restrict__ selected,
                                float* __restrict__ out) {
    __shared__ float As[16][132];   // 132-float row stride: banks m*4 mod 64 distinct

    const int t = threadIdx.x;              // 0..255
    const int rowbase = blockIdx.x * 16;

    // Stage 16 gathered rows (16 x 128 floats) cooperatively, float4 loads.
    for (int q = t; q < 16 * 32; q += 256) {
        const int r  = q >> 5;
        const int c4 = q & 31;
        const int src = selected[rowbase + r];
        const float4 v = ((const float4*)(x + (size_t)src * C_))[c4];
        *(float4*)&As[r][c4 * 4] = v;       // row stride 528B (16B multiple)
    }
    __syncthreads();

    const int wave = t >> 5;
    const int lane = t & 31;
    const int g16  = lane >> 4;             // lane group 0/1
    const int mr   = lane & 15;
    const int col  = wave * 16 + mr;        // this lane's output column

    v8f acc = {};
    for (int kk = 0; kk < 32; ++kk) {
        const int kb = kk * 4 + 2 * g16;    // K index of VGPR0 for this lane
        const v2f a = *(const v2f*)&As[mr][kb];
        v2f b;
        b.x = w_t[(size_t)kb * C_ + col];
        b.y = w_t[(size_t)(kb + 1) * C_ + col];
        acc = __builtin_amdgcn_wmma_f32_16x16x4_f32(
            /*neg_a=*/false, a, /*neg_b=*/false, b,
            /*c_mod=*/(short)0, acc, /*reuse_a=*/false, /*reuse_b=*/false);
    }

    const float bias = b_t[col];
    #pragma unroll
    for (int r = 0; r < 8; ++r) {
        const int m = r + 8 * g16;
        out[(size_t)(rowbase + m) * C_ + col] = acc[r] + bias;
    }
}

// ---------------------------------------------------------------------------
// Kernel 4: remap edges through node_id_map; masked edges -> (-1,-1), mask 0/1.
// ---------------------------------------------------------------------------
__global__ void hap_edge_kernel(const int* __restrict__ ei,
                                const int* __restrict__ node_map,
                                float* __restrict__ out_edge,
                                float* __restrict__ out_mask) {
    const int e = blockIdx.x * 256 + threadIdx.x;
    if (e >= E_) return;
    const int r = node_map[ei[e]];
    const int c = node_map[ei[E_ + e]];
    const bool m = (r >= 0) && (c >= 0);
    out_edge[e]      = m ? (float)r : -1.0f;
    out_edge[E_ + e] = m ? (float)c : -1.0f;
    out_mask[e]      = m ? 1.0f : 0.0f;
}

// ---------------------------------------------------------------------------
// Launch: inputs in setup_inputs() order:
//   0:x 1:edge_index 2:batch 3:w_score 4:b_score 5:w_t 6:b_t
// Outputs concatenated as float: pooled_x | edge_index_pool | new_batch | mask
// ---------------------------------------------------------------------------
extern "C" void kernel_launch(void* const* d_in, const int* in_sizes, int n_in,
                              void* d_out, int out_size, void* d_ws, size_t ws_size,
                              hipStream_t stream) {
    const float* x        = (const float*)d_in[0];
    const int*   ei       = (const int*)d_in[1];
    const float* w_score  = (const float*)d_in[3];
    const float* b_score  = (const float*)d_in[4];
    const float* w_t      = (const float*)d_in[5];
    const float* b_t      = (const float*)d_in[6];

    // workspace layout
    char* ws = (char*)d_ws;
    float* scores   = (float*)ws;                                   // TOTAL_ floats
    int*   selected = (int*)(ws + (size_t)TOTAL_ * sizeof(float));  // G_*K_ ints
    int*   node_map = (int*)(ws + (size_t)TOTAL_ * sizeof(float)
                                + (size_t)G_ * K_ * sizeof(int));   // TOTAL_ ints

    // output layout (floats)
    float* out        = (float*)d_out;
    float* out_pooled = out;                                        // G_*K_*C_
    float* out_edge   = out + (size_t)G_ * K_ * C_;                 // 2*E_
    float* out_batch  = out_edge + (size_t)2 * E_;                  // G_*K_
    float* out_mask   = out_batch + (size_t)G_ * K_;                // E_

    hap_score_kernel<<<TOTAL_ / 8, 256, 0, stream>>>(x, w_score, b_score,
                                                     scores, node_map);
    hap_topk_kernel<<<G_, 512, 0, stream>>>(scores, selected, node_map, out_batch);
    hap_gemm_kernel<<<(G_ * K_) / 16, 256, 0, stream>>>(x, w_t, b_t, selected,
                                                        out_pooled);
    hap_edge_kernel<<<(E_ + 255) / 256, 256, 0, stream>>>(ei, node_map,
                                                          out_edge, out_mask);
}